// GPSDepthAttentionLayer_5437428596935
// MI455X (gfx1250) — compile-verified
//
#include <hip/hip_runtime.h>
#include <hip/hip_bf16.h>
#include <math.h>

typedef float v2f __attribute__((ext_vector_type(2)));
typedef float v8f __attribute__((ext_vector_type(8)));

#define FIN  256
#define FOUT 256
#define HID  16
#define FEAT 17            // HID + 1
#define FC   51            // 3 * FEAT

// ---------------------------------------------------------------- utilities
__global__ void k_fill(float* __restrict__ p, float v, int n) {
    int i = blockIdx.x * blockDim.x + threadIdx.x;
    if (i < n) p[i] = v;
}

// deg (init 1.0 for self loops) over adj[1]; cnt_adj over adj[0]; cnt_e over edges[0]
__global__ void k_counts(const int* __restrict__ adj, const int* __restrict__ edges,
                         float* __restrict__ deg, float* __restrict__ cnt_adj,
                         float* __restrict__ cnt_e, int E) {
    int e = blockIdx.x * blockDim.x + threadIdx.x;
    if (e >= E) return;
    unsafeAtomicAdd(&deg[adj[E + e]], 1.0f);
    unsafeAtomicAdd(&cnt_adj[adj[e]], 1.0f);
    unsafeAtomicAdd(&cnt_e[edges[e]], 1.0f);
}

// deg -> dinv in place; clamp counts to >= 1
__global__ void k_norm(float* __restrict__ deg_dinv, float* __restrict__ cnt_adj,
                       float* __restrict__ cnt_e, int n) {
    int i = blockIdx.x * blockDim.x + threadIdx.x;
    if (i >= n) return;
    float d = deg_dinv[i];
    deg_dinv[i] = (d > 0.0f) ? rsqrtf(d) : 0.0f;
    cnt_adj[i] = fmaxf(cnt_adj[i], 1.0f);
    cnt_e[i]   = fmaxf(cnt_e[i], 1.0f);
}

// ------------------------------------------------ GEMM1: xw = (lpw .* x) @ W
// One wave (32 lanes) computes a 16x64 output panel with V_WMMA_F32_16X16X4_F32.
// A 16x4 f32: lane L holds row M=L%16, k = (L/16)*2 + {0,1} in a[0],a[1].
// B 4x16 f32: lane L holds col N=L%16, k = (L/16)*2 + {0,1} in b[0],b[1].
// C/D 16x16:  VGPR v, lane L -> row = v + (L/16)*8, col = L%16.
__global__ __launch_bounds__(32) void k_gemm1(const float* __restrict__ x,
                                              const float* __restrict__ lpw,
                                              const float* __restrict__ W,
                                              float* __restrict__ xw, int n) {
    const int lane = threadIdx.x;
    const int m0   = blockIdx.y << 4;
    const int n0   = blockIdx.x << 6;
    const int half = lane >> 4;          // 0 or 1
    const int kb   = half << 1;          // 0 or 2
    const int ncol = lane & 15;
    int mrow = m0 + (lane & 15);
    if (mrow >= n) mrow = n - 1;         // clamp loads; stores guarded below
    const float scale = lpw[mrow];
    const float* __restrict__ xrow = x + (size_t)mrow * FIN;

    v8f c0 = {}; v8f c1 = {}; v8f c2 = {}; v8f c3 = {};
    for (int k = 0; k < FIN; k += 4) {
        v2f a;
        a.x = scale * xrow[k + kb];
        a.y = scale * xrow[k + kb + 1];
        const float* __restrict__ w0 = W + (size_t)(k + kb) * FOUT + n0 + ncol;
        const float* __restrict__ w1 = w0 + FOUT;
        v2f b;
        b.x = w0[0];  b.y = w1[0];
        c0 = __builtin_amdgcn_wmma_f32_16x16x4_f32(false, a, false, b, (short)0, c0, false, false);
        b.x = w0[16]; b.y = w1[16];
        c1 = __builtin_amdgcn_wmma_f32_16x16x4_f32(false, a, false, b, (short)0, c1, false, false);
        b.x = w0[32]; b.y = w1[32];
        c2 = __builtin_amdgcn_wmma_f32_16x16x4_f32(false, a, false, b, (short)0, c2, false, false);
        b.x = w0[48]; b.y = w1[48];
        c3 = __builtin_amdgcn_wmma_f32_16x16x4_f32(false, a, false, b, (short)0, c3, false, false);
    }
    const int rbase = m0 + (half << 3);
#pragma unroll
    for (int v = 0; v < 8; ++v) {
        int r = rbase + v;
        if (r < n) {
            float* __restrict__ o = xw + (size_t)r * FOUT + n0 + ncol;
            o[0] = c0[v]; o[16] = c1[v]; o[32] = c2[v]; o[48] = c3[v];
        }
    }
}

// accum[i,f] = dinv[i]^2 * xw[i,f]   (self-loop term seeds the accumulator)
__global__ void k_gcn_init(const float* __restrict__ xw, const float* __restrict__ dinv,
                           float* __restrict__ accum, int n) {
    int idx = blockIdx.x * blockDim.x + threadIdx.x;
    if (idx >= n * FOUT) return;
    int i = idx >> 8;
    float di = dinv[i];
    accum[idx] = di * di * xw[idx];
}

// one wave per edge: accum[dst,:] += dinv[src]*dinv[dst] * xw[src,:]
__global__ void k_gcn_scatter(const int* __restrict__ adj, const float* __restrict__ xw,
                              const float* __restrict__ dinv, float* __restrict__ accum,
                              int E) {
    int wid  = (blockIdx.x * blockDim.x + threadIdx.x) >> 5;
    int lane = threadIdx.x & 31;
    if (wid >= E) return;
    int s = adj[wid], d = adj[E + wid];
    float w = dinv[s] * dinv[d];
    const float* __restrict__ xs = xw + (size_t)s * FOUT;
    float* __restrict__ od = accum + (size_t)d * FOUT;
#pragma unroll
    for (int f = lane; f < FOUT; f += 32)
        unsafeAtomicAdd(&od[f], w * xs[f]);
}

// final_h = aggr*(accum + conv_b) + (1-aggr)*input    (in place on accum==out)
__global__ void k_gcn_final(float* __restrict__ out, const float* __restrict__ input,
                            const float* __restrict__ conv_b, const float* __restrict__ aggr,
                            int n) {
    int idx = blockIdx.x * blockDim.x + threadIdx.x;
    if (idx >= n * FOUT) return;
    float a = aggr[0];
    int f = idx & (FOUT - 1);
    out[idx] = a * (out[idx] + conv_b[f]) + (1.0f - a) * input[idx];
}

// ------------------------------------------------ GEMM2: hm = X @ W2 + b2  [n,16]
__global__ __launch_bounds__(32) void k_gemm2(const float* __restrict__ X,
                                              const float* __restrict__ W2,
                                              const float* __restrict__ b2,
                                              float* __restrict__ hm, int n) {
    const int lane = threadIdx.x;
    const int m0   = blockIdx.x << 4;
    const int half = lane >> 4;
    const int kb   = half << 1;
    const int ncol = lane & 15;
    int mrow = m0 + (lane & 15);
    if (mrow >= n) mrow = n - 1;
    const float* __restrict__ xrow = X + (size_t)mrow * FIN;

    v8f c = {};
    for (int k = 0; k < FIN; k += 4) {
        v2f a;
        a.x = xrow[k + kb];
        a.y = xrow[k + kb + 1];
        v2f b;
        b.x = W2[(size_t)(k + kb) * HID + ncol];
        b.y = W2[(size_t)(k + kb + 1) * HID + ncol];
        c = __builtin_amdgcn_wmma_f32_16x16x4_f32(false, a, false, b, (short)0, c, false, false);
    }
    const int rbase = m0 + (half << 3);
    const float bias = b2[ncol];
#pragma unroll
    for (int v = 0; v < 8; ++v) {
        int r = rbase + v;
        if (r < n) hm[(size_t)r * HID + ncol] = c[v] + bias;
    }
}

// acc16[adj0[e], f] += hm[adj1[e], f]
__global__ void k_avg_scatter(const int* __restrict__ adj, const float* __restrict__ hm,
                              float* __restrict__ acc16, int E) {
    int idx = blockIdx.x * blockDim.x + threadIdx.x;
    if (idx >= E * HID) return;
    int e = idx >> 4, f = idx & 15;
    unsafeAtomicAdd(&acc16[(size_t)adj[e] * HID + f], hm[(size_t)adj[E + e] * HID + f]);
}

__global__ void k_avg_div(const float* __restrict__ acc16, const float* __restrict__ cnt_adj,
                          float* __restrict__ avg, int n) {
    int idx = blockIdx.x * blockDim.x + threadIdx.x;
    if (idx >= n * HID) return;
    avg[idx] = acc16[idx] / cnt_adj[idx >> 4];
}

// per-edge pairwise distance, accumulated per source node
__global__ void k_dist(const int* __restrict__ edges, const float* __restrict__ avg,
                       const float* __restrict__ hm, float* __restrict__ distacc, int E) {
    int e = blockIdx.x * blockDim.x + threadIdx.x;
    if (e >= E) return;
    int s = edges[e], d = edges[E + e];
    const float* __restrict__ L = avg + (size_t)s * HID;
    const float* __restrict__ R = hm + (size_t)d * HID;
    float sum = 0.0f;
#pragma unroll
    for (int f = 0; f < HID; ++f) {
        float t = L[f] - R[f] + 1e-6f;
        sum += t * t;
    }
    unsafeAtomicAdd(&distacc[s], sqrtf(sum));
}

__global__ void k_div_n(const float* __restrict__ acc, const float* __restrict__ cnt,
                        float* __restrict__ out, int n) {
    int i = blockIdx.x * blockDim.x + threadIdx.x;
    if (i < n) out[i] = acc[i] / cnt[i];
}

// ---------------------------------------------------------- per-edge MLPs
__device__ __forceinline__ float mlp_eval(const float* fc, const float* sW1,
                                          const float* sB1, const float* sW2, float sB2) {
    float o = sB2;
#pragma unroll
    for (int j = 0; j < HID; ++j) {
        float t = sB1[j];
        for (int i = 0; i < FC; ++i) t += fc[i] * sW1[i * HID + j];
        o += tanhf(t) * sW2[j];
    }
    return 1.0f / (1.0f + expf(-o));
}

__global__ void k_factor0(const int* __restrict__ edges, const float* __restrict__ hm0,
                          const float* __restrict__ d20,
                          const float* __restrict__ W1, const float* __restrict__ B1,
                          const float* __restrict__ W2v, const float* __restrict__ B2v,
                          float* __restrict__ factor0, int E) {
    __shared__ float sW1[FC * HID];
    __shared__ float sB1[HID];
    __shared__ float sW2[HID];
    __shared__ float sB2;
    for (int t = threadIdx.x; t < FC * HID; t += blockDim.x) sW1[t] = W1[t];
    if (threadIdx.x < HID) { sB1[threadIdx.x] = B1[threadIdx.x]; sW2[threadIdx.x] = W2v[threadIdx.x]; }
    if (threadIdx.x == 0) sB2 = B2v[0];
    __syncthreads();
    int e = blockIdx.x * blockDim.x + threadIdx.x;
    if (e >= E) return;
    int s = edges[e], d = edges[E + e];
    const float* __restrict__ hs = hm0 + (size_t)s * HID;
    const float* __restrict__ hd = hm0 + (size_t)d * HID;
    float fc[FC];
#pragma unroll
    for (int f = 0; f < HID; ++f) {
        fc[f] = hs[f];
        fc[FEAT + f] = hd[f];
        fc[2 * FEAT + f] = fabsf(hd[f] - hs[f]);
    }
    fc[HID] = d20[s];
    fc[FEAT + HID] = d20[d];
    fc[2 * FEAT + HID] = fabsf(d20[d] - d20[s]);
    factor0[e] = mlp_eval(fc, sW1, sB1, sW2, sB2);
}

// pass 1: s1/d1 from feat1 + att_bias; h_diff recomputed from pass-0 features
__global__ void k_factor1(const int* __restrict__ edges, const float* __restrict__ hm1,
                          const float* __restrict__ d21, const float* __restrict__ hm0,
                          const float* __restrict__ d20, const float* __restrict__ att_bias,
                          const float* __restrict__ W1, const float* __restrict__ B1,
                          const float* __restrict__ W2v, const float* __restrict__ B2v,
                          float* __restrict__ r1acc, int E) {
    __shared__ float sW1[FC * HID];
    __shared__ float sB1[HID];
    __shared__ float sW2[HID];
    __shared__ float sAB[FEAT];
    __shared__ float sB2;
    for (int t = threadIdx.x; t < FC * HID; t += blockDim.x) sW1[t] = W1[t];
    if (threadIdx.x < HID) { sB1[threadIdx.x] = B1[threadIdx.x]; sW2[threadIdx.x] = W2v[threadIdx.x]; }
    if (threadIdx.x < FEAT) sAB[threadIdx.x] = att_bias[threadIdx.x];
    if (threadIdx.x == 0) sB2 = B2v[0];
    __syncthreads();
    int e = blockIdx.x * blockDim.x + threadIdx.x;
    if (e >= E) return;
    int s = edges[e], d = edges[E + e];
    const float* __restrict__ hs1 = hm1 + (size_t)s * HID;
    const float* __restrict__ hd1 = hm1 + (size_t)d * HID;
    const float* __restrict__ hs0 = hm0 + (size_t)s * HID;
    const float* __restrict__ hd0 = hm0 + (size_t)d * HID;
    float fc[FC];
#pragma unroll
    for (int f = 0; f < HID; ++f) {
        fc[f] = hs1[f] + sAB[f];
        fc[FEAT + f] = hd1[f] + sAB[f];
        fc[2 * FEAT + f] = fabsf(hd0[f] - hs0[f]);
    }
    fc[HID] = d21[s] + sAB[HID];
    fc[FEAT + HID] = d21[d] + sAB[HID];
    fc[2 * FEAT + HID] = fabsf(d20[d] - d20[s]);
    unsafeAtomicAdd(&r1acc[s], mlp_eval(fc, sW1, sB1, sW2, sB2));
}

__global__ void k_r1_div(float* __restrict__ r1, const float* __restrict__ cnt_e, int n) {
    int i = blockIdx.x * blockDim.x + threadIdx.x;
    if (i < n) r1[i] = r1[i] / cnt_e[i];
}

__global__ void k_r2_scatter(const int* __restrict__ edges, const float* __restrict__ factor0,
                             const float* __restrict__ r1, float* __restrict__ r2, int E) {
    int e = blockIdx.x * blockDim.x + threadIdx.x;
    if (e >= E) return;
    unsafeAtomicAdd(&r2[edges[e]], factor0[e] * r1[edges[E + e]]);
}

// ---------------------------------------------------------------- launcher
extern "C" void kernel_launch(void* const* d_in, const int* in_sizes, int n_in,
                              void* d_out, int out_size, void* d_ws, size_t ws_size,
                              hipStream_t stream) {
    const float* input    = (const float*)d_in[0];
    const int*   adj      = (const int*)  d_in[1];
    const float* aggr     = (const float*)d_in[2];
    const int*   edges    = (const int*)  d_in[3];
    const float* lpw      = (const float*)d_in[7];
    const float* conv_W   = (const float*)d_in[8];
    const float* conv_b   = (const float*)d_in[9];
    const float* W2       = (const float*)d_in[10];
    const float* b2       = (const float*)d_in[11];
    const float* att_bias = (const float*)d_in[12];
    const float* lf1_W    = (const float*)d_in[13];
    const float* lf1_b    = (const float*)d_in[14];
    const float* lf2_W    = (const float*)d_in[15];
    const float* lf2_b    = (const float*)d_in[16];

    const int n = in_sizes[0] / FIN;
    const int E = in_sizes[1] / 2;

    float* ws = (float*)d_ws;
    float* xw      = ws; ws += (size_t)n * FOUT;
    float* dinv    = ws; ws += n;          // holds deg, then dinv
    float* cnt_adj = ws; ws += n;
    float* cnt_e   = ws; ws += n;
    float* hm0     = ws; ws += (size_t)n * HID;
    float* hm1     = ws; ws += (size_t)n * HID;
    float* acc16   = ws; ws += (size_t)n * HID;
    float* avg     = ws; ws += (size_t)n * HID;
    float* distacc = ws; ws += n;
    float* d20     = ws; ws += n;
    float* d21     = ws; ws += n;
    float* factor0 = ws; ws += E;
    float* r1      = ws; ws += n;
    float* r2      = ws; ws += n;

    float* final_h = (float*)d_out;
    float* out2    = final_h + (size_t)n * FOUT;

    const int TPB = 256;
    const int gN     = (n + TPB - 1) / TPB;
    const int gE     = (E + TPB - 1) / TPB;
    const int gNF    = (int)(((size_t)n * FOUT + TPB - 1) / TPB);
    const int gNH    = (int)(((size_t)n * HID + TPB - 1) / TPB);
    const int gEH    = (int)(((size_t)E * HID + TPB - 1) / TPB);
    const int gEwave = (E * 32 + TPB - 1) / TPB;   // one wave per edge

    // zero accumulators
    hipMemsetAsync(cnt_adj, 0, (size_t)n * sizeof(float), stream);
    hipMemsetAsync(cnt_e,   0, (size_t)n * sizeof(float), stream);
    hipMemsetAsync(acc16,   0, (size_t)n * HID * sizeof(float), stream);
    hipMemsetAsync(distacc, 0, (size_t)n * sizeof(float), stream);
    hipMemsetAsync(r1,      0, (size_t)n * sizeof(float), stream);
    hipMemsetAsync(r2,      0, (size_t)n * sizeof(float), stream);
    k_fill<<<gN, TPB, 0, stream>>>(dinv, 1.0f, n);   // deg starts at 1 (self loop)

    // degrees / counts / normalization
    k_counts<<<gE, TPB, 0, stream>>>(adj, edges, dinv, cnt_adj, cnt_e, E);
    k_norm<<<gN, TPB, 0, stream>>>(dinv, cnt_adj, cnt_e, n);

    // xw = (lpw .* input) @ conv_W   (WMMA f32)
    dim3 g1(FOUT / 64, (n + 15) / 16);
    k_gemm1<<<g1, 32, 0, stream>>>(input, lpw, conv_W, xw, n);

    // GCN aggregation: self-loop seed + edge scatter + combine
    k_gcn_init<<<gNF, TPB, 0, stream>>>(xw, dinv, final_h, n);
    k_gcn_scatter<<<gEwave, TPB, 0, stream>>>(adj, xw, dinv, final_h, E);
    k_gcn_final<<<gNF, TPB, 0, stream>>>(final_h, input, conv_b, aggr, n);

    // ---- pass 0: features from final_h
    k_gemm2<<<(n + 15) / 16, 32, 0, stream>>>(final_h, W2, b2, hm0, n);
    k_avg_scatter<<<gEH, TPB, 0, stream>>>(adj, hm0, acc16, E);
    k_avg_div<<<gNH, TPB, 0, stream>>>(acc16, cnt_adj, avg, n);
    k_dist<<<gE, TPB, 0, stream>>>(edges, avg, hm0, distacc, E);
    k_div_n<<<gN, TPB, 0, stream>>>(distacc, cnt_e, d20, n);
    k_factor0<<<gE, TPB, 0, stream>>>(edges, hm0, d20, lf1_W, lf1_b, lf2_W, lf2_b, factor0, E);

    // ---- pass 1: features from raw input (h_diff reused from pass 0)
    hipMemsetAsync(acc16,   0, (size_t)n * HID * sizeof(float), stream);
    hipMemsetAsync(distacc, 0, (size_t)n * sizeof(float), stream);
    k_gemm2<<<(n + 15) / 16, 32, 0, stream>>>(input, W2, b2, hm1, n);
    k_avg_scatter<<<gEH, TPB, 0, stream>>>(adj, hm1, acc16, E);
    k_avg_div<<<gNH, TPB, 0, stream>>>(acc16, cnt_adj, avg, n);
    k_dist<<<gE, TPB, 0, stream>>>(edges, avg, hm1, distacc, E);
    k_div_n<<<gN, TPB, 0, stream>>>(distacc, cnt_e, d21, n);
    k_factor1<<<gE, TPB, 0, stream>>>(edges, hm1, d21, hm0, d20, att_bias,
                                      lf1_W, lf1_b, lf2_W, lf2_b, r1, E);

    // res_1hop / res_2hop
    k_r1_div<<<gN, TPB, 0, stream>>>(r1, cnt_e, n);
    k_r2_scatter<<<gE, TPB, 0, stream>>>(edges, factor0, r1, r2, E);
    k_div_n<<<gN, TPB, 0, stream>>>(r2, cnt_e, out2, n);
}